// DirichletProcess_VI_76991583748649
// MI455X (gfx1250) — compile-verified
//
#include <hip/hip_runtime.h>
#include <math.h>

typedef float v2f __attribute__((ext_vector_type(2)));
typedef float v8f __attribute__((ext_vector_type(8)));
typedef int   v4i __attribute__((ext_vector_type(4)));
typedef __attribute__((address_space(1))) v4i* as1_v4i;
typedef __attribute__((address_space(3))) v4i* as3_v4i;

#define TCOMP 32
#define DIM   1024
#define BATCH 4096
#define KCHUNK 128
#define LDSP  132   // padded LDS pitch (floats): 4-bank row skew, conflict-free b64 reads

#if __has_builtin(__builtin_amdgcn_global_load_async_to_lds_b128)
#define HAVE_ASYNC_LDS 1
#else
#define HAVE_ASYNC_LDS 0
#endif

// workspace layout (float offsets)
#define WS_INV   0                       // inv_var        [32*1024]
#define WS_MUI   (32*1024)               // mu*inv_var     [32*1024]
#define WS_CT    (2*32*1024)             // c_t            [32]
#define WS_NTG   (WS_CT + 32)            // N_t_gaussian   [32]
#define WS_NTPI  (WS_NTG + 32)           // N_t_pi         [32]
#define WS_KLG   (WS_NTPI + 32)          // kl_gaussian    [BATCH*32]
#define WS_LPI   (WS_KLG + BATCH*TCOMP)  // log_pi         [BATCH*32]

__global__ void initk(float* ws, float* out) {
    int tid = threadIdx.x;
    if (tid < 64) ws[WS_NTG + tid] = 0.0f;   // covers NTG[32] + NTPI[32]
    if (tid == 0) out[0] = 0.0f;
}

// Per-component precompute: inv_var, mu*inv_var, c_t = sum(mu^2 * inv_var)
__global__ __launch_bounds__(256) void phase0(const float* __restrict__ mu,
                                              const float* __restrict__ rho,
                                              float* __restrict__ ws) {
    int t = blockIdx.x;
    int tid = threadIdx.x;
    float c = 0.0f;
    for (int i = 0; i < 4; ++i) {
        int d = i * 256 + tid;
        float r = rho[t * DIM + d];
        float s = (r > 20.0f) ? r : log1pf(__expf(r));   // softplus
        float inv = 1.0f / (s * s);
        float m = mu[t * DIM + d];
        ws[WS_INV + t * DIM + d] = inv;
        ws[WS_MUI + t * DIM + d] = m * inv;
        c += m * m * inv;
    }
    __shared__ float red[256];
    red[tid] = c;
    __syncthreads();
    for (int s = 128; s > 0; s >>= 1) {
        if (tid < s) red[tid] += red[tid + s];
        __syncthreads();
    }
    if (tid == 0) ws[WS_CT + t] = red[0];
}

// Stick-breaking log_pi via wave32 prefix scan; column sums into N_t_pi
__global__ __launch_bounds__(256) void logpi_k(const float* __restrict__ beta,
                                               float* __restrict__ ws) {
    int tid  = threadIdx.x;
    int lane = tid & 31;
    int w    = tid >> 5;
    int row  = blockIdx.x * 8 + w;
    float b   = beta[row * TCOMP + lane];
    float l1m = logf(1.0f - b);
    float sc = l1m;                              // inclusive scan of log(1-beta)
    for (int d = 1; d < 32; d <<= 1) {
        float o = __shfl_up(sc, d, 32);
        if (lane >= d) sc += o;
    }
    float lp = logf(b) + (sc - l1m);             // exclusive prefix
    ws[WS_LPI + row * TCOMP + lane] = lp;

    __shared__ float col[32];
    if (tid < 32) col[tid] = 0.0f;
    __syncthreads();
    atomicAdd(&col[lane], lp);
    __syncthreads();
    if (tid < 32) atomicAdd(&ws[WS_NTPI + tid], col[tid]);
}

// Main fused dual-GEMM: kl_gaussian = 0.5*(D - x^2@inv_var^T + 2*x@muinv^T - c_t)
// 4 waves: wave = (gemm<<1)|tcol; each wave owns one 16x16 f32 WMMA accumulator.
// x tile staged via double-buffered GLOBAL_LOAD_ASYNC_TO_LDS_B128 (ASYNCcnt).
__global__ __launch_bounds__(128) void gemm_klg(const float* __restrict__ x,
                                                float* __restrict__ ws) {
    __shared__ float xs[2][16 * LDSP];
    __shared__ float res[2][16][33];
    __shared__ float col[32];

    int tid  = threadIdx.x;
    int lane = tid & 31;
    int wave = tid >> 5;
    int gemm = wave >> 1;       // 0: A = x^2, W = inv_var   1: A = x, W = mu*inv_var
    int tcol = wave & 1;        // which half of the 32 components
    int tile = blockIdx.x;

    const float* W = ws + (gemm ? WS_MUI : WS_INV) + (size_t)(tcol * 16 + (lane & 15)) * DIM;
    int koff = 2 * (lane >> 4);         // lane-half selects K pair {0,1} vs {2,3}
    int arow = lane & 15;
    const float* xbase = x + (size_t)tile * 16 * DIM;

    v8f acc = {0.f, 0.f, 0.f, 0.f, 0.f, 0.f, 0.f, 0.f};

    // per-thread slot mapping: 512 float4 slots covering a 16 x 128 chunk
    int slotR[4], slotC[4];
    #pragma unroll
    for (int i = 0; i < 4; ++i) {
        int slot = i * 128 + tid;
        slotR[i] = slot >> 5;           // 32 float4 per row
        slotC[i] = (slot & 31) * 4;
    }

#if HAVE_ASYNC_LDS
    // prologue: async-stage chunk 0 into buffer 0
    #pragma unroll
    for (int i = 0; i < 4; ++i)
        __builtin_amdgcn_global_load_async_to_lds_b128(
            (as1_v4i)(xbase + slotR[i] * DIM + slotC[i]),
            (as3_v4i)(&xs[0][slotR[i] * LDSP + slotC[i]]), 0, 0);
#endif

    for (int kc = 0; kc < DIM; kc += KCHUNK) {
        int p = (kc / KCHUNK) & 1;
#if HAVE_ASYNC_LDS
        if (kc + KCHUNK < DIM) {
            // async-stage next chunk into the other buffer (overlaps WMMA below)
            #pragma unroll
            for (int i = 0; i < 4; ++i)
                __builtin_amdgcn_global_load_async_to_lds_b128(
                    (as1_v4i)(xbase + slotR[i] * DIM + kc + KCHUNK + slotC[i]),
                    (as3_v4i)(&xs[1 - p][slotR[i] * LDSP + slotC[i]]), 0, 0);
            asm volatile("s_wait_asynccnt 0x4" ::: "memory");  // chunk kc complete
        } else {
            asm volatile("s_wait_asynccnt 0x0" ::: "memory");
        }
#else
        #pragma unroll
        for (int i = 0; i < 4; ++i) {
            float4 v = *(const float4*)(xbase + slotR[i] * DIM + kc + slotC[i]);
            *(float4*)(&xs[p][slotR[i] * LDSP + slotC[i]]) = v;
        }
#endif
        __syncthreads();   // async LDS writes visible to all waves

        if (kc + 2 * KCHUNK < DIM)  // warm L2 beyond the double buffer
            __builtin_prefetch(xbase + (tid >> 3) * DIM + kc + 2 * KCHUNK + (tid & 7) * 16, 0, 0);

        const float* xsp = &xs[p][arow * LDSP + koff];
        if (gemm == 0) {            // wave-uniform branch: squaring variant
            #pragma unroll 8
            for (int ks = 0; ks < KCHUNK; ks += 4) {
                v2f a = *(const v2f*)(xsp + ks);
                a.x = a.x * a.x; a.y = a.y * a.y;
                v2f b = *(const v2f*)(W + kc + ks + koff);
                acc = __builtin_amdgcn_wmma_f32_16x16x4_f32(
                    false, a, false, b, (short)0, acc, false, false);
            }
        } else {
            #pragma unroll 8
            for (int ks = 0; ks < KCHUNK; ks += 4) {
                v2f a = *(const v2f*)(xsp + ks);
                v2f b = *(const v2f*)(W + kc + ks + koff);
                acc = __builtin_amdgcn_wmma_f32_16x16x4_f32(
                    false, a, false, b, (short)0, acc, false, false);
            }
        }
        __syncthreads();
    }

    // C/D layout: VGPR v -> row v (lanes 0-15) / row v+8 (lanes 16-31), col = lane&15
    {
        int rbase = (lane >> 4) << 3;
        int c = lane & 15;
        #pragma unroll
        for (int v = 0; v < 8; ++v)
            res[gemm][rbase + v][tcol * 16 + c] = acc[v];
    }
    if (tid < 32) col[tid] = 0.0f;
    __syncthreads();

    float* klg = ws + WS_KLG + (size_t)tile * 16 * TCOMP;
    for (int i = 0; i < 4; ++i) {
        int e = i * 128 + tid;
        int r = e >> 5, t = e & 31;
        float g1 = res[0][r][t];
        float g2 = res[1][r][t];
        float v = 0.5f * (float)DIM - 0.5f * g1 + g2 - 0.5f * ws[WS_CT + t];
        klg[r * TCOMP + t] = v;
        atomicAdd(&col[t], v);
    }
    __syncthreads();
    if (tid < 32) atomicAdd(&ws[WS_NTG + tid], col[tid]);
}

// mix, per-row softmax over T=32 lanes, weighted sum, mean over batch
__global__ __launch_bounds__(256) void phase2(const float* __restrict__ ws,
                                              float* __restrict__ out) {
    int tid  = threadIdx.x;
    int lane = tid & 31;
    int w    = tid >> 5;
    int row  = blockIdx.x * 8 + w;

    float ntg = ws[WS_NTG + lane];
    float ntp = ws[WS_NTPI + lane];
    float mix = ntp / (ntg + ntp);

    float klg = ws[WS_KLG + row * TCOMP + lane];
    float lpi = ws[WS_LPI + row * TCOMP + lane];
    float kl  = mix * klg + (1.0f - mix) * lpi;

    float m = kl;
    for (int d = 16; d > 0; d >>= 1) m = fmaxf(m, __shfl_xor(m, d, 32));
    float e = __expf(kl - m);
    float s = e;
    for (int d = 16; d > 0; d >>= 1) s += __shfl_xor(s, d, 32);
    float v = (e / s) * (mix * klg);
    for (int d = 16; d > 0; d >>= 1) v += __shfl_xor(v, d, 32);

    __shared__ float red[8];
    if (lane == 0) red[w] = v;
    __syncthreads();
    if (tid == 0) {
        float acc = 0.0f;
        for (int i = 0; i < 8; ++i) acc += red[i];
        atomicAdd(out, acc * (1.0f / (float)BATCH));
    }
}

extern "C" void kernel_launch(void* const* d_in, const int* in_sizes, int n_in,
                              void* d_out, int out_size, void* d_ws, size_t ws_size,
                              hipStream_t stream) {
    (void)in_sizes; (void)n_in; (void)out_size; (void)ws_size;
    const float* x    = (const float*)d_in[0];
    const float* mu   = (const float*)d_in[1];
    const float* rho  = (const float*)d_in[2];
    const float* beta = (const float*)d_in[3];
    float* out = (float*)d_out;
    float* ws  = (float*)d_ws;

    initk   <<<1,            64, 0, stream>>>(ws, out);
    phase0  <<<TCOMP,       256, 0, stream>>>(mu, rho, ws);
    logpi_k <<<BATCH / 8,   256, 0, stream>>>(beta, ws);
    gemm_klg<<<BATCH / 16,  128, 0, stream>>>(x, ws);
    phase2  <<<BATCH / 8,   256, 0, stream>>>(ws, out);
}